// myGATConv_4784593568246
// MI455X (gfx1250) — compile-verified
//
#include <hip/hip_runtime.h>
#include <hip/hip_bf16.h>
#include <math.h>

#define N_NODES 10000
#define DEG 32
#define IN_FEATS 256
#define HEADS 8
#define OUT_FEATS 64
#define HD (HEADS * OUT_FEATS) /* 512 */
#define NEG_SLOPE 0.2f

typedef __attribute__((ext_vector_type(2))) float v2f;
typedef __attribute__((ext_vector_type(8))) float v8f;

// ---------------------------------------------------------------------------
// Kernel 1: ft = feat @ W  ([10000x256] x [256x512]) via V_WMMA_F32_16X16X4_F32
// One wave -> 16 rows x 64 cols (4 accumulators); K stepped by 4.
// 10000 rows = 625 tiles of 16 (exact), 512 cols = 8 waves x 64.
// ---------------------------------------------------------------------------
__global__ __launch_bounds__(256) void gat_gemm_wmma(
    const float* __restrict__ feat, const float* __restrict__ W,
    float* __restrict__ ft)
{
    const int lane  = threadIdx.x & 31;
    const int wave  = threadIdx.x >> 5;        // 0..7 -> 64-column group
    const int m0    = blockIdx.x * 16;         // row tile
    const int nb    = wave * 64;
    const int mrow  = m0 + (lane & 15);        // A: row per lane
    const int khalf = (lane >> 4) * 2;         // A/B: K pair 0 or 2
    const int ncol  = lane & 15;               // B/D: column per lane

    v8f acc0 = {}, acc1 = {}, acc2 = {}, acc3 = {};
    const float* arow = feat + (size_t)mrow * IN_FEATS;

    for (int k0 = 0; k0 < IN_FEATS; k0 += 4) {
        const int kb = k0 + khalf;
        // A fragment: lane holds feat[mrow, kb], feat[mrow, kb+1]  (8B aligned)
        v2f a = *(const v2f*)(arow + kb);
        // B fragments: W[kb, n] / W[kb+1, n] for 4 column tiles
        const float* w0 = W + (size_t)kb * HD + nb + ncol;
        const float* w1 = w0 + HD;
        v2f b0, b1, b2, b3;
        b0.x = w0[0];  b0.y = w1[0];
        b1.x = w0[16]; b1.y = w1[16];
        b2.x = w0[32]; b2.y = w1[32];
        b3.x = w0[48]; b3.y = w1[48];
        acc0 = __builtin_amdgcn_wmma_f32_16x16x4_f32(false, a, false, b0, (short)0, acc0, false, false);
        acc1 = __builtin_amdgcn_wmma_f32_16x16x4_f32(false, a, false, b1, (short)0, acc1, false, false);
        acc2 = __builtin_amdgcn_wmma_f32_16x16x4_f32(false, a, false, b2, (short)0, acc2, false, false);
        acc3 = __builtin_amdgcn_wmma_f32_16x16x4_f32(false, a, false, b3, (short)0, acc3, false, false);
    }

    // C/D layout: VGPR r -> row m0 + r + 8*(lane>>4), col = lane&15
    const int rbase = m0 + ((lane >> 4) << 3);
#pragma unroll
    for (int r = 0; r < 8; ++r) {
        float* dp = ft + (size_t)(rbase + r) * HD + nb + ncol;
        dp[0]  = acc0[r];
        dp[16] = acc1[r];
        dp[32] = acc2[r];
        dp[48] = acc3[r];
    }
}

// ---------------------------------------------------------------------------
// Kernel 2: el[n,h] = <ft[n,h,:], attn_l[h,:]>, er likewise.
// One block per node, wave h handles head h; 64-elem dot via 2 loads/lane +
// wave32 shuffle reduction.
// ---------------------------------------------------------------------------
__global__ __launch_bounds__(256) void gat_logits(
    const float* __restrict__ ft, const float* __restrict__ attn_l,
    const float* __restrict__ attn_r, float* __restrict__ el,
    float* __restrict__ er)
{
    const int n    = blockIdx.x;
    const int h    = threadIdx.x >> 5;
    const int lane = threadIdx.x & 31;

    const float* f  = ft + (size_t)n * HD + h * OUT_FEATS;
    const float* al = attn_l + h * OUT_FEATS;
    const float* ar = attn_r + h * OUT_FEATS;

    const float f0 = f[lane], f1 = f[lane + 32];
    float sl = f0 * al[lane] + f1 * al[lane + 32];
    float sr = f0 * ar[lane] + f1 * ar[lane + 32];
#pragma unroll
    for (int m = 16; m > 0; m >>= 1) {
        sl += __shfl_xor(sl, m, 32);
        sr += __shfl_xor(sr, m, 32);
    }
    if (lane == 0) {
        el[n * HEADS + h] = sl;
        er[n * HEADS + h] = sr;
    }
}

// ---------------------------------------------------------------------------
// Kernel 3: per-node edge logits + LeakyReLU + softmax over 32 edges + agg.
// dst = repeat(arange(N), 32) -> node n owns edges [n*32, n*32+32).
// Block(256) per node: 32 edges x 8 heads; wave h does head-h softmax.
// ---------------------------------------------------------------------------
__global__ __launch_bounds__(256) void gat_edge_softmax_agg(
    const float* __restrict__ ft, const float* __restrict__ el,
    const float* __restrict__ er, const int* __restrict__ src,
    const int* __restrict__ dst, float* __restrict__ rst,
    float* __restrict__ a_out)
{
    __shared__ int   s_src[DEG];
    __shared__ float s_e[DEG][HEADS];
    __shared__ float s_a[DEG][HEADS];

    const int    n     = blockIdx.x;
    const int    t     = threadIdx.x;
    const long   ebase = (long)n * DEG;

    if (t < DEG) s_src[t] = src[ebase + t];
    __syncthreads();

    // edge logits + LeakyReLU: thread t -> (edge j, head h)
    {
        const int j = t >> 3, h = t & 7;
        const int dn = dst[ebase + j];   // == n for this dataset
        float e = el[s_src[j] * HEADS + h] + er[dn * HEADS + h];
        s_e[j][h] = (e > 0.f) ? e : NEG_SLOPE * e;
    }
    __syncthreads();

    // per-head softmax across the 32 edges (wave h, lane j)
    {
        const int h = t >> 5, j = t & 31;
        float e  = s_e[j][h];
        float mx = e;
#pragma unroll
        for (int m = 16; m > 0; m >>= 1) mx = fmaxf(mx, __shfl_xor(mx, m, 32));
        float ex  = __expf(e - mx);
        float sum = ex;
#pragma unroll
        for (int m = 16; m > 0; m >>= 1) sum += __shfl_xor(sum, m, 32);
        const float a = ex / sum;
        s_a[j][h] = a;
        a_out[(ebase + j) * HEADS + h] = a;  // [E, H, 1] flat
    }
    __syncthreads();

    // aggregation: rst[n,h,d] = sum_j a[j,h] * ft[src_j, h, d]
#pragma unroll
    for (int rep = 0; rep < 2; ++rep) {
        const int idx = t + rep * 256;        // 512 outputs per node
        const int h = idx >> 6, d = idx & 63; // d contiguous -> coalesced L2 gathers
        float acc = 0.f;
#pragma unroll 8
        for (int j = 0; j < DEG; ++j)
            acc += s_a[j][h] * ft[(size_t)s_src[j] * HD + h * OUT_FEATS + d];
        rst[(size_t)n * HD + idx] = acc;
    }
}

// ---------------------------------------------------------------------------
extern "C" void kernel_launch(void* const* d_in, const int* in_sizes, int n_in,
                              void* d_out, int out_size, void* d_ws, size_t ws_size,
                              hipStream_t stream)
{
    const float* feat   = (const float*)d_in[0];
    const float* W      = (const float*)d_in[1];
    const float* attn_l = (const float*)d_in[2];
    const float* attn_r = (const float*)d_in[3];
    const int*   src    = (const int*)d_in[4];
    const int*   dst    = (const int*)d_in[5];

    float* out_rst = (float*)d_out;                       // [N, H, D]
    float* out_a   = out_rst + (size_t)N_NODES * HD;      // [E, H, 1]

    float* ft = (float*)d_ws;                             // [N, 512]  ~20.5 MB
    float* el = ft + (size_t)N_NODES * HD;                // [N, 8]
    float* er = el + (size_t)N_NODES * HEADS;             // [N, 8]

    gat_gemm_wmma<<<N_NODES / 16, 256, 0, stream>>>(feat, W, ft);
    gat_logits<<<N_NODES, 256, 0, stream>>>(ft, attn_l, attn_r, el, er);
    gat_edge_softmax_agg<<<N_NODES, 256, 0, stream>>>(ft, el, er, src, dst,
                                                      out_rst, out_a);
}